// SemiConv2d_66219805769826
// MI455X (gfx1250) — compile-verified
//
#include <hip/hip_runtime.h>
#include <stdint.h>

// Max-plus 5x5 dilation over (16,64,256,256) f32, 'same' padding with -inf.
// out[b,c,i,j] = max_{u,v in 0..4} x[b,c, i+u-2, j+v-2] + k[u,v]
//
// CDNA5 (gfx1250) strategy: stage a (TH+4)x(TW+4) input tile into LDS with
// per-lane async global->LDS copies (ASYNCcnt), -inf for out-of-image halo,
// then a fully unrolled v_add_f32 / v_max_num_f32 register micro-kernel with
// 128-bit LDS reads and 128-bit global stores.

#define KH 5
#define KW 5
#define HALO 2
#define HW 256            // H == W == 256
#define TW 64             // output tile width
#define TH 32             // output tile height
#define LCOLS (TW + 4)    // 68 valid staged columns
#define LSTRIDE (TW + 8)  // 72-float row stride -> 288 B, 16-B aligned rows
#define LROWS (TH + 4)    // 36 staged rows
#define NCELLS (LROWS * LCOLS)
#define NTHREADS 256

__global__ __launch_bounds__(NTHREADS) void dilate5x5_kernel(
    const float* __restrict__ in, const float* __restrict__ kern,
    float* __restrict__ out) {
  __shared__ float tile[LROWS * LSTRIDE];

  const int tid = threadIdx.x;
  const int x0 = blockIdx.x * TW;
  const int y0 = blockIdx.y * TH;
  const uint32_t plane_off = (uint32_t)blockIdx.z * (uint32_t)(HW * HW);

  // ---- Stage halo tile into LDS via CDNA5 async global->LDS copies ----
  const uint32_t lds_base = (uint32_t)(uintptr_t)(&tile[0]);
  for (int t = tid; t < NCELLS; t += NTHREADS) {
    const int r = t / LCOLS;
    const int c = t - r * LCOLS;
    const int gy = y0 + r - HALO;
    const int gx = x0 + c - HALO;
    const uint32_t lds_addr = lds_base + (uint32_t)(r * LSTRIDE + c) * 4u;
    if ((unsigned)gy < HW && (unsigned)gx < HW) {
      const uint32_t goff =
          (plane_off + (uint32_t)gy * HW + (uint32_t)gx) * 4u;
      // vdst = per-lane LDS byte address, vaddr = 32-bit byte offset,
      // saddr = 64-bit base (GVS mode). Tracked by ASYNCcnt.
      asm volatile("global_load_async_to_lds_b32 %0, %1, %2"
                   :
                   : "v"(lds_addr), "v"(goff), "s"(in)
                   : "memory");
    } else {
      tile[r * LSTRIDE + c] = -__builtin_inff();  // -inf padding
    }
  }
  asm volatile("s_wait_asynccnt 0" ::: "memory");
  __syncthreads();

  // ---- Kernel taps: uniform address -> scalar loads ----
  float kk[KH * KW];
#pragma unroll
  for (int i = 0; i < KH * KW; ++i) kk[i] = kern[i];

  const int qx = (tid & 15) * 4;  // quad column within tile (0,4,...,60)
  const int ty = tid >> 4;        // 0..15

#pragma unroll
  for (int half = 0; half < 2; ++half) {
    const int orow = ty + half * 16;  // 0..31
    float acc0 = -__builtin_inff();
    float acc1 = acc0, acc2 = acc0, acc3 = acc0;

#pragma unroll
    for (int u = 0; u < KH; ++u) {
      const float* rp = &tile[(orow + u) * LSTRIDE + qx];  // 16-B aligned
      const float4 A = *(const float4*)(rp);
      const float4 B = *(const float4*)(rp + 4);
      float w[8] = {A.x, A.y, A.z, A.w, B.x, B.y, B.z, B.w};
#pragma unroll
      for (int v = 0; v < KW; ++v) {
        const float kv = kk[u * KW + v];
        acc0 = fmaxf(acc0, w[v + 0] + kv);
        acc1 = fmaxf(acc1, w[v + 1] + kv);
        acc2 = fmaxf(acc2, w[v + 2] + kv);
        acc3 = fmaxf(acc3, w[v + 3] + kv);
      }
    }

    const uint32_t oidx =
        plane_off + (uint32_t)(y0 + orow) * HW + (uint32_t)(x0 + qx);
    float4 o;
    o.x = acc0; o.y = acc1; o.z = acc2; o.w = acc3;
    *(float4*)&out[oidx] = o;  // 128-bit store
  }
}

extern "C" void kernel_launch(void* const* d_in, const int* in_sizes, int n_in,
                              void* d_out, int out_size, void* d_ws,
                              size_t ws_size, hipStream_t stream) {
  (void)in_sizes; (void)n_in; (void)d_ws; (void)ws_size; (void)out_size;
  const float* in = (const float*)d_in[0];    // (16,64,256,256) f32
  const float* kern = (const float*)d_in[1];  // (5,5) f32
  float* out = (float*)d_out;

  dim3 grid(HW / TW, HW / TH, 16 * 64);  // 4 x 8 x 1024
  dim3 block(NTHREADS);
  dilate5x5_kernel<<<grid, block, 0, stream>>>(in, kern, out);
}